// WildcatPool2d_7937099563299
// MI455X (gfx1250) — compile-verified
//
#include <hip/hip_runtime.h>

// WildcatPool2d for x[64,128,64,64] f32 -> out[64,128] f32.
// One 256-thread block per row (4096 floats). Row staged into LDS with CDNA5
// async global->LDS B128 copies (ASYNCcnt), keys held in registers, exact
// top-k/bottom-k (k=819) thresholds found by a fused dual radix-select with
// per-wave LDS histograms and wave32 shuffle scans.

#define ROW_N    4096
#define THREADS  256
#define NWAVES   (THREADS / 32)      // 8 waves
#define ELEMS    (ROW_N / THREADS)   // 16 per thread
#define K_SEL    819
#define ALPHA_F  0.7f
#define HSTRIDE  512                 // per-wave: 256 bins sel0 + 256 bins sel1

__device__ __forceinline__ unsigned float_to_key(unsigned u) {
    return (u & 0x80000000u) ? ~u : (u | 0x80000000u);
}
__device__ __forceinline__ float key_to_float(unsigned k) {
    unsigned u = (k & 0x80000000u) ? (k ^ 0x80000000u) : ~k;
    return __uint_as_float(u);
}

__global__ __launch_bounds__(THREADS) void wildcat_pool2d_kernel(
        const float* __restrict__ x, float* __restrict__ out) {
    // 16 KB buffer: async staging first, then 8 per-wave histograms [8][512]
    __shared__ __align__(16) unsigned smem[ROW_N];
    __shared__ unsigned s_wtot[2 * NWAVES];   // scan wave totals (sel0, sel1)
    __shared__ unsigned s_prefix[2];
    __shared__ unsigned s_krem[2];
    __shared__ float    s_wsum[2 * NWAVES];

    const int t = threadIdx.x;
    const int wave = t >> 5;
    const int lane = t & 31;
    const long long row = blockIdx.x;
    const float* rowp = x + row * (long long)ROW_N;

    // ---- Stage row into LDS: async global->LDS, non-temporal (single use) ----
    {
        unsigned lds_base = (unsigned)(size_t)(&smem[0]);  // low 32 bits = LDS offset
        #pragma unroll
        for (int it = 0; it < 4; ++it) {
            unsigned off = (unsigned)(it * (THREADS * 16) + t * 16);
            unsigned ldsaddr = lds_base + off;
            unsigned long long gaddr =
                (unsigned long long)(const void*)((const char*)rowp + off);
            asm volatile("global_load_async_to_lds_b128 %0, %1, off th:TH_LOAD_NT scope:SCOPE_DEV"
                         :: "v"(ldsaddr), "v"(gaddr) : "memory");
        }
        asm volatile("s_wait_asynccnt 0" ::: "memory");
    }
    __syncthreads();

    // ---- Keys to registers (strided, order irrelevant for selection/sums) ----
    unsigned myk[ELEMS];
    #pragma unroll
    for (int e = 0; e < ELEMS; ++e) {
        myk[e] = float_to_key(smem[t + e * THREADS]);
    }
    __syncthreads();   // smem now reusable as histograms

    // ---- Fused dual radix select: sel0 = k-th smallest, sel1 = k-th largest ----
    unsigned prefix0 = 0, prefix1 = 0;
    unsigned krem0 = K_SEL, krem1 = K_SEL;

    for (int pass = 0; pass < 4; ++pass) {
        const int shift = 24 - pass * 8;

        // clear all 4096 histogram slots with B128 stores
        {
            uint4 z = make_uint4(0u, 0u, 0u, 0u);
            uint4* p4 = (uint4*)smem;
            #pragma unroll
            for (int i = 0; i < 4; ++i) p4[t + i * THREADS] = z;
        }
        __syncthreads();

        unsigned* h = &smem[wave * HSTRIDE];
        if (pass == 0) {
            // top-byte histogram; sel1's pass-0 counts are the mirror image
            #pragma unroll
            for (int e = 0; e < ELEMS; ++e) {
                atomicAdd(&h[myk[e] >> 24], 1u);
            }
        } else {
            const unsigned maskhi = 0xFFFFFFFFu << (shift + 8);
            #pragma unroll
            for (int e = 0; e < ELEMS; ++e) {
                unsigned k0 = myk[e];
                if ((k0 & maskhi) == prefix0)
                    atomicAdd(&h[(k0 >> shift) & 255u], 1u);
                unsigned k1 = ~k0;
                if ((k1 & maskhi) == prefix1)
                    atomicAdd(&h[256u + ((k1 >> shift) & 255u)], 1u);
            }
        }
        __syncthreads();

        // gather this thread's bin counts across the 8 per-wave histograms
        unsigned c0 = 0, c1 = 0;
        #pragma unroll
        for (int w = 0; w < NWAVES; ++w) {
            c0 += smem[w * HSTRIDE + t];
            c1 += (pass == 0) ? smem[w * HSTRIDE + (255 - t)]
                              : smem[w * HSTRIDE + 256 + t];
        }

        // wave32 inclusive shuffle scans (no barriers)
        unsigned v0 = c0, v1 = c1;
        #pragma unroll
        for (int o = 1; o < 32; o <<= 1) {
            unsigned n0 = __shfl_up(v0, o, 32);
            unsigned n1 = __shfl_up(v1, o, 32);
            if (lane >= o) { v0 += n0; v1 += n1; }
        }
        if (lane == 31) { s_wtot[wave] = v0; s_wtot[NWAVES + wave] = v1; }
        __syncthreads();

        unsigned off0 = 0, off1 = 0;
        for (int w = 0; w < wave; ++w) {
            off0 += s_wtot[w];
            off1 += s_wtot[NWAVES + w];
        }
        unsigned incl0 = v0 + off0, excl0 = incl0 - c0;
        unsigned incl1 = v1 + off1, excl1 = incl1 - c1;
        if (excl0 < krem0 && krem0 <= incl0) {
            s_prefix[0] = prefix0 | ((unsigned)t << shift);
            s_krem[0]   = krem0 - excl0;
        }
        if (excl1 < krem1 && krem1 <= incl1) {
            s_prefix[1] = prefix1 | ((unsigned)t << shift);
            s_krem[1]   = krem1 - excl1;
        }
        __syncthreads();
        prefix0 = s_prefix[0]; krem0 = s_krem[0];
        prefix1 = s_prefix[1]; krem1 = s_krem[1];
        __syncthreads();
    }
    // prefix0/1 = exact threshold keys (working space); krem = copies at threshold

    // ---- Strict-side sums + krem * threshold value ----
    float ps0 = 0.0f, ps1 = 0.0f;
    #pragma unroll
    for (int e = 0; e < ELEMS; ++e) {
        unsigned k0 = myk[e];
        float v = key_to_float(k0);
        if (k0 < prefix0)  ps0 += v;    // strictly below bottom threshold
        if (~k0 < prefix1) ps1 += v;    // strictly above top threshold
    }
    #pragma unroll
    for (int o = 16; o > 0; o >>= 1) {
        ps0 += __shfl_down(ps0, o, 32);
        ps1 += __shfl_down(ps1, o, 32);
    }
    if (lane == 0) { s_wsum[wave] = ps0; s_wsum[NWAVES + wave] = ps1; }
    __syncthreads();

    if (t == 0) {
        float bot = 0.0f, top = 0.0f;
        #pragma unroll
        for (int w = 0; w < NWAVES; ++w) {
            bot += s_wsum[w];
            top += s_wsum[NWAVES + w];
        }
        bot += (float)krem0 * key_to_float(prefix0);
        top += (float)krem1 * key_to_float(~prefix1);
        float bot_mean = bot / (float)K_SEL;
        float top_mean = top / (float)K_SEL;
        out[row] = (top_mean + ALPHA_F * bot_mean) * 0.5f;
    }
}

extern "C" void kernel_launch(void* const* d_in, const int* in_sizes, int n_in,
                              void* d_out, int out_size, void* d_ws, size_t ws_size,
                              hipStream_t stream) {
    (void)d_ws; (void)ws_size;
    const float* x = (const float*)d_in[0];
    float* out = (float*)d_out;
    int rows = out_size;                 // B*C = 8192
    if (rows <= 0 && n_in > 0) rows = in_sizes[0] / ROW_N;
    wildcat_pool2d_kernel<<<dim3((unsigned)rows), dim3(THREADS), 0, stream>>>(x, out);
}